// PathConvModule_32126355374224
// MI455X (gfx1250) — compile-verified
//
#include <hip/hip_runtime.h>
#include <hip/hip_bf16.h>

typedef __attribute__((ext_vector_type(2))) float v2f;
typedef __attribute__((ext_vector_type(8))) float v8f;
typedef __attribute__((ext_vector_type(4))) unsigned int u32x4;
typedef __attribute__((ext_vector_type(4))) int i32x4;
typedef __attribute__((ext_vector_type(8))) int i32x8;

#define DD 64
#define NN 64
#define BB 4

#if defined(__has_builtin)
#if __has_builtin(__builtin_amdgcn_tensor_load_to_lds) && __has_builtin(__builtin_amdgcn_s_wait_tensorcnt)
#define HAVE_TDM 1
#endif
#endif

// ============ pack W (K x 64, row-major) into WMMA-B pair layout ============
// WB[p*64+col] = { W[2p][col], W[2p+1][col] }  -> one b64 load per B fragment.
__global__ void k_packW(const float* __restrict__ W, v2f* __restrict__ WB, int K) {
  int g = blockIdx.x * blockDim.x + threadIdx.x;
  int tot = (K >> 1) * DD;
  if (g < tot) {
    int p = g >> 6, col = g & 63;
    v2f t;
    t.x = W[(2 * p) * DD + col];
    t.y = W[(2 * p + 1) * DD + col];
    WB[g] = t;
  }
}

// ============ K1: Z = path_emb @ W_hz + b_hz  (M=16384, K=64, N=64) ============
__global__ __launch_bounds__(32)
void k_gemm_hz(const float* __restrict__ X, const v2f* __restrict__ WB,
               const float* __restrict__ bias, float* __restrict__ Z) {
  const int lane = threadIdx.x;
  const int r0   = blockIdx.x * 16;
  const int mrow = lane & 15;
  const int kofs = (lane >> 4) * 2;        // A: lanes 0-15 hold K={0,1}, 16-31 hold K={2,3}
  const float* xrow = X + (size_t)(r0 + mrow) * DD;
  v8f acc[4] = {};
  for (int k = 0; k < DD; k += 4) {
    v2f a = *(const v2f*)(xrow + k + kofs);
    const v2f* wrow = WB + (size_t)((k + kofs) >> 1) * DD;
#pragma unroll
    for (int nt = 0; nt < 4; ++nt) {
      v2f bm = wrow[nt * 16 + mrow];
      acc[nt] = __builtin_amdgcn_wmma_f32_16x16x4_f32(false, a, false, bm,
                                                      (short)0, acc[nt], false, false);
    }
  }
  const int rhalf = (lane >> 4) * 8;        // C/D: VGPR v -> row v (+8 for hi lanes)
#pragma unroll
  for (int nt = 0; nt < 4; ++nt) {
    const int col = nt * 16 + mrow;
    const float bv = bias[col];
#pragma unroll
    for (int v = 0; v < 8; ++v)
      Z[(size_t)(r0 + rhalf + v) * DD + col] = acc[nt][v] + bv;
  }
}

// ===== K2: tropical (max,+) conv:  ZC[b,i,j,d] = max( max_k Z[b,i,k,d]+Z[b,k,j,d], Z[b,i,j,d] )
// Block per (b,i). The contiguous 16KB row-slab Z[b,i,:,:] is DMA'd into LDS by the
// Tensor Data Mover (wave 0 issues; TENSORcnt waited; then workgroup barrier).
__global__ __launch_bounds__(256)
void k_maxplus(const float* __restrict__ Z, float* __restrict__ ZC) {
  __shared__ float zi[NN * DD];             // only LDS object -> lds_addr 0
  const int bi = blockIdx.x;
  const int b = bi >> 6, i = bi & 63;
  const int tid = threadIdx.x;
  const float* zslab = Z + (size_t)b * NN * NN * DD;
  const float* zrow  = zslab + (size_t)i * NN * DD;   // 4096 contiguous floats

#ifdef HAVE_TDM
  if (tid < 32) {                           // one wave drives the TDM DMA
    unsigned long long ga = (unsigned long long)(uintptr_t)zrow;
    u32x4 g0 = {};
    g0[0] = 1u;                                           // count=1 (valid user D#)
    g0[1] = 0u;                                           // lds_addr = 0
    g0[2] = (unsigned)(ga & 0xffffffffull);               // global_addr[31:0]
    g0[3] = (unsigned)((ga >> 32) & 0x1ffffffull)         // global_addr[56:32]
            | (2u << 30);                                 // type = 2 ("image")
    i32x8 g1 = {};
    g1[0] = (int)(2u << 16);                              // data_size=2 -> 4 bytes
    g1[1] = (int)((4096u & 0xffffu) << 16);               // tensor_dim0 lo16
    g1[2] = (int)(((4096u >> 16) & 0xffffu)               // tensor_dim0 hi16
                  | (1u << 16));                          // tensor_dim1 lo16 = 1
    g1[3] = (int)((4096u & 0xffffu) << 16);               // tile_dim0 = 4096
    g1[4] = 1;                                            // tile_dim1 = 1
    g1[5] = 4096;                                         // tensor_dim0_stride lo32
    g1[6] = 0;
    g1[7] = 0;
    i32x4 g2 = {};
    i32x4 g3 = {};
#if __clang_major__ >= 23
    i32x8 g4 = {};
    __builtin_amdgcn_tensor_load_to_lds(g0, g1, g2, g3, g4, 0);
#else
    __builtin_amdgcn_tensor_load_to_lds(g0, g1, g2, g3, 0);
#endif
    __builtin_amdgcn_s_wait_tensorcnt((short)0);          // s_wait_tensorcnt 0
  }
  __syncthreads();
  // The TDM engine wrote zi behind the compiler's back; escape the array into
  // opaque asm with a memory clobber so the LDS reads below cannot be proven
  // undef and DCE'd (this cost us the whole max-plus loop last round).
  asm volatile("" :: "v"(&zi[0]) : "memory");
#else
  for (int idx = tid; idx < NN * DD; idx += 256) zi[idx] = zrow[idx];
  __syncthreads();
#endif

  const int d  = tid & 63;                  // lane->bank 1:1, conflict free
  const int jg = tid >> 6;
  for (int jj = 0; jj < 16; ++jj) {
    const int j = jg * 16 + jj;
    const float* zcol = zslab + (size_t)j * DD + d;       // Z[b,k,j,d], k-stride 4096
    float acc = -3.402823466e38f;
#pragma unroll 4
    for (int k = 0; k < NN; ++k)
      acc = fmaxf(acc, zi[k * DD + d] + zcol[(size_t)k * NN * DD]);
    acc = fmaxf(acc, zi[j * DD + d]);                     // path_aggr 'max' with z
    ZC[((size_t)(b * NN + i) * NN + j) * DD + d] = acc;
  }
}

// ===== K3: out_path = relu([n_i | zc | n_j] @ W_zz + b_zz) + path_emb  (M=16384, K=192, N=64)
__global__ __launch_bounds__(32)
void k_gemm_zz(const float* __restrict__ ne, const float* __restrict__ ZC,
               const v2f* __restrict__ WB, const float* __restrict__ bias,
               const float* __restrict__ path_emb, float* __restrict__ out_path) {
  const int lane = threadIdx.x;
  const int r0   = blockIdx.x * 16;
  const int mrow = lane & 15;
  const int r    = r0 + mrow;
  const int bb = r >> 12, ii = (r >> 6) & 63, jj = r & 63;
  const int kofs = (lane >> 4) * 2;
  const float* ai = ne + (size_t)(bb * NN + ii) * DD;   // k in [0,64)
  const float* az = ZC + (size_t)r * DD;                // k in [64,128)
  const float* aj = ne + (size_t)(bb * NN + jj) * DD;   // k in [128,192)
  v8f acc[4] = {};
  for (int k = 0; k < 3 * DD; k += 4) {
    const int kk = k + kofs;                 // float2 never straddles a 4-aligned boundary
    v2f a;
    if (kk < 64)       a = *(const v2f*)(ai + kk);
    else if (kk < 128) a = *(const v2f*)(az + kk - 64);
    else               a = *(const v2f*)(aj + kk - 128);
    const v2f* wrow = WB + (size_t)(kk >> 1) * DD;
#pragma unroll
    for (int nt = 0; nt < 4; ++nt) {
      v2f bm = wrow[nt * 16 + mrow];
      acc[nt] = __builtin_amdgcn_wmma_f32_16x16x4_f32(false, a, false, bm,
                                                      (short)0, acc[nt], false, false);
    }
  }
  const int rhalf = (lane >> 4) * 8;
#pragma unroll
  for (int nt = 0; nt < 4; ++nt) {
    const int col = nt * 16 + mrow;
    const float bv = bias[col];
#pragma unroll
    for (int v = 0; v < 8; ++v) {
      const size_t idx = (size_t)(r0 + rhalf + v) * DD + col;
      float val = acc[nt][v] + bv;
      val = val > 0.f ? val : 0.f;           // relu
      out_path[idx] = val + path_emb[idx];   // residual
    }
  }
}

// ============ K4: segment_sum via float atomics ============
__global__ void k_zero(float* __restrict__ p, int nElts) {
  int g = blockIdx.x * blockDim.x + threadIdx.x;
  if (g < nElts) p[g] = 0.f;
}
__global__ void k_scatter(const int* __restrict__ eidx, const float* __restrict__ ne,
                          float* __restrict__ aggr, int E) {
  int g = blockIdx.x * blockDim.x + threadIdx.x;
  int e = g >> 6, d = g & 63;
  if (e < E) {
    int s = eidx[e];        // src row
    int t = eidx[E + e];    // dst row
    atomicAdd(&aggr[(size_t)t * DD + d], ne[(size_t)s * DD + d]);
  }
}

// ===== K5: out_node = relu([node_emb | aggr] @ W_nm + b_nm) + node_emb  (M=256, K=128, N=64)
__global__ __launch_bounds__(32)
void k_gemm_nm(const float* __restrict__ ne, const float* __restrict__ aggr,
               const v2f* __restrict__ WB, const float* __restrict__ bias,
               float* __restrict__ out_node) {
  const int lane = threadIdx.x;
  const int r0   = blockIdx.x * 16;
  const int mrow = lane & 15;
  const int r    = r0 + mrow;
  const int kofs = (lane >> 4) * 2;
  const float* a0 = ne   + (size_t)r * DD;
  const float* a1 = aggr + (size_t)r * DD;
  v8f acc[4] = {};
  for (int k = 0; k < 2 * DD; k += 4) {
    const int kk = k + kofs;
    v2f a = (kk < 64) ? *(const v2f*)(a0 + kk) : *(const v2f*)(a1 + kk - 64);
    const v2f* wrow = WB + (size_t)(kk >> 1) * DD;
#pragma unroll
    for (int nt = 0; nt < 4; ++nt) {
      v2f bm = wrow[nt * 16 + mrow];
      acc[nt] = __builtin_amdgcn_wmma_f32_16x16x4_f32(false, a, false, bm,
                                                      (short)0, acc[nt], false, false);
    }
  }
  const int rhalf = (lane >> 4) * 8;
#pragma unroll
  for (int nt = 0; nt < 4; ++nt) {
    const int col = nt * 16 + mrow;
    const float bv = bias[col];
#pragma unroll
    for (int v = 0; v < 8; ++v) {
      const size_t idx = (size_t)(r0 + rhalf + v) * DD + col;
      float val = acc[nt][v] + bv;
      val = val > 0.f ? val : 0.f;
      out_node[idx] = val + ne[idx];
    }
  }
}

extern "C" void kernel_launch(void* const* d_in, const int* in_sizes, int n_in,
                              void* d_out, int out_size, void* d_ws, size_t ws_size,
                              hipStream_t stream) {
  const float* node_emb = (const float*)d_in[0];   // (256, 64)
  const float* path_emb = (const float*)d_in[1];   // (4, 64, 64, 64)
  const int*   edge_idx = (const int*)d_in[2];     // (2, 4096)
  const float* W_hz = (const float*)d_in[3];
  const float* b_hz = (const float*)d_in[4];
  const float* W_zz = (const float*)d_in[5];
  const float* b_zz = (const float*)d_in[6];
  const float* W_nm = (const float*)d_in[7];
  const float* b_nm = (const float*)d_in[8];

  float* out       = (float*)d_out;
  float* out_node  = out;                               // 256*64
  float* out_path  = out + 256 * DD;                    // 4*64*64*64

  const size_t slab = (size_t)BB * NN * NN * DD;        // 1,048,576 floats
  float* Z    = (float*)d_ws;                           // z = path_emb@W_hz + b
  float* ZC   = Z + slab;                               // max-plus result
  float* aggr = ZC + slab;                              // 256*64 segment sums
  v2f*   WBhz = (v2f*)(aggr + 256 * DD);                // 32*64 pairs
  v2f*   WBzz = WBhz + 32 * DD;                         // 96*64 pairs
  v2f*   WBnm = WBzz + 96 * DD;                         // 64*64 pairs

  const int Mpath = BB * NN * NN;                       // 16384
  const int E = 4096;

  // repack weights into WMMA-B pair layout (once per launch; ~98KB total)
  k_packW<<<(32 * DD + 255) / 256, 256, 0, stream>>>(W_hz, WBhz, DD);
  k_packW<<<(96 * DD + 255) / 256, 256, 0, stream>>>(W_zz, WBzz, 3 * DD);
  k_packW<<<(64 * DD + 255) / 256, 256, 0, stream>>>(W_nm, WBnm, 2 * DD);

  // path branch
  k_gemm_hz<<<Mpath / 16, 32, 0, stream>>>(path_emb, WBhz, b_hz, Z);
  k_maxplus<<<BB * NN, 256, 0, stream>>>(Z, ZC);
  k_gemm_zz<<<Mpath / 16, 32, 0, stream>>>(node_emb, ZC, WBzz, b_zz, path_emb, out_path);

  // node branch
  k_zero<<<(256 * DD + 255) / 256, 256, 0, stream>>>(aggr, 256 * DD);
  k_scatter<<<(E * DD + 255) / 256, 256, 0, stream>>>(edge_idx, node_emb, aggr, E);
  k_gemm_nm<<<256 / 16, 32, 0, stream>>>(node_emb, aggr, WBnm, b_nm, out_node);
}